// SBM_61658550501979
// MI455X (gfx1250) — compile-verified
//
#include <hip/hip_runtime.h>

#define V 2048
#define NB 16
#define NSTEP 10
#define EPS 1e-5f
#define NTILE (V / 16)   // 128 row-tiles per batch

typedef float v2f __attribute__((ext_vector_type(2)));
typedef float v8f __attribute__((ext_vector_type(8)));

// ---------------------------------------------------------------------------
// x0[b,i] = sum_j A[b,j,i]   (column sums; coalesced across lanes, 8 streams)
// grid (V/64, NB), block 64
// ---------------------------------------------------------------------------
__global__ void sbm_colsum(const float* __restrict__ A, float* __restrict__ x) {
    const int b = blockIdx.y;
    const int i = blockIdx.x * 64 + threadIdx.x;
    const float* __restrict__ Ab = A + (size_t)b * V * V;
    float acc[8] = {0.f, 0.f, 0.f, 0.f, 0.f, 0.f, 0.f, 0.f};
    for (int j = 0; j < V; j += 8) {
#pragma unroll
        for (int u = 0; u < 8; ++u)
            acc[u] += Ab[(size_t)(j + u) * V + i];
    }
    float s = ((acc[0] + acc[1]) + (acc[2] + acc[3])) +
              ((acc[4] + acc[5]) + (acc[6] + acc[7]));
    x[b * V + i] = s;
}

#define WMMA_F32(a, b, c) \
    __builtin_amdgcn_wmma_f32_16x16x4_f32(false, (a), false, (b), (short)0, (c), false, false)

// ---------------------------------------------------------------------------
// One step: s = A[b] @ x ; y = relu(alpha*x + beta*s) ; per-tile partial sums.
// WMMA f32 16x16x4: A-op = 16x4 fp32 tile of A, B-op = x[j0..j0+3] broadcast
// into all 16 columns -> every D column holds the same partial matvec.
//
// A-op layout (ISA 7.12.2, 32-bit A 16x4): lane (m = l&15, h = l>>4):
//   VGPR0 = A[m][j0+2h], VGPR1 = A[m][j0+2h+1]        -> one b64 load.
// B-op layout (mirrors documented B tables): lane half h:
//   VGPR0 = x[j0+2h], VGPR1 = x[j0+2h+1]              -> one b64 load.
// D layout (ISA): lane n, VGPR r = D[M = r + 8h][n].
//
// Software-pipelined: next 16-column slab is loaded before the current
// slab's WMMAs issue, so waits are partial (loadcnt<=8) and ~16 loads stay
// in flight per wave instead of 2.
//
// grid (NTILE/4, NB), block 128 (4 waves, one 16-row tile per wave)
// ---------------------------------------------------------------------------
__global__ void __launch_bounds__(128)
sbm_step(const float* __restrict__ A, const float* __restrict__ x,
         const float* __restrict__ alpha, const float* __restrict__ beta, int k,
         float* __restrict__ y, float* __restrict__ psum,
         float* __restrict__ psumsq) {
    const int b    = blockIdx.y;
    const int lane = threadIdx.x & 31;
    const int wave = threadIdx.x >> 5;
    const int tile = blockIdx.x * 4 + wave;  // 0..127
    const int i0   = tile * 16;
    const int m    = lane & 15;
    const int h    = lane >> 4;

    const float* __restrict__ Ab   = A + (size_t)b * V * V;
    const float* __restrict__ Arow = Ab + (size_t)(i0 + m) * V + 2 * h;
    const float* __restrict__ xb   = x + b * V;
    const float* __restrict__ xk   = xb + 2 * h;

    v8f acc0 = {}; v8f acc1 = {}; v8f acc2 = {}; v8f acc3 = {};

    // ---- prologue: load first slab ----
    v2f ca0 = *(const v2f*)(Arow + 0);
    v2f ca1 = *(const v2f*)(Arow + 4);
    v2f ca2 = *(const v2f*)(Arow + 8);
    v2f ca3 = *(const v2f*)(Arow + 12);
    v2f cb0 = *(const v2f*)(xk + 0);
    v2f cb1 = *(const v2f*)(xk + 4);
    v2f cb2 = *(const v2f*)(xk + 8);
    v2f cb3 = *(const v2f*)(xk + 12);

#pragma unroll 2
    for (int j0 = 0; j0 < V - 16; j0 += 16) {
        const float* __restrict__ An = Arow + j0 + 16;
        const float* __restrict__ xn = xk + j0 + 16;
        // issue next-slab loads before consuming current slab
        v2f na0 = *(const v2f*)(An + 0);
        v2f na1 = *(const v2f*)(An + 4);
        v2f na2 = *(const v2f*)(An + 8);
        v2f na3 = *(const v2f*)(An + 12);
        v2f nb0 = *(const v2f*)(xn + 0);
        v2f nb1 = *(const v2f*)(xn + 4);
        v2f nb2 = *(const v2f*)(xn + 8);
        v2f nb3 = *(const v2f*)(xn + 12);
        __builtin_prefetch(An + 512, 0, 0);  // global_prefetch_b8, ~2KB ahead

        acc0 = WMMA_F32(ca0, cb0, acc0);
        acc1 = WMMA_F32(ca1, cb1, acc1);
        acc2 = WMMA_F32(ca2, cb2, acc2);
        acc3 = WMMA_F32(ca3, cb3, acc3);

        ca0 = na0; ca1 = na1; ca2 = na2; ca3 = na3;
        cb0 = nb0; cb1 = nb1; cb2 = nb2; cb3 = nb3;
    }
    // ---- epilogue: last slab ----
    acc0 = WMMA_F32(ca0, cb0, acc0);
    acc1 = WMMA_F32(ca1, cb1, acc1);
    acc2 = WMMA_F32(ca2, cb2, acc2);
    acc3 = WMMA_F32(ca3, cb3, acc3);

    v8f acc = (acc0 + acc1) + (acc2 + acc3);

    // Extract s[i0 + M]: all D columns identical; lane n==0 of each half owns
    // rows M = r + 8h in VGPR r.
    __shared__ float sred[4 * 16];
    if (m == 0) {
#pragma unroll
        for (int r = 0; r < 8; ++r) sred[wave * 16 + 8 * h + r] = acc[r];
    }
    __syncthreads();
    float s = sred[wave * 16 + m];

    const float a_k = alpha[k];
    const float b_k = beta[k];
    float xv = xb[i0 + m];
    float yv = fmaxf(a_k * xv + b_k * s, 0.0f);
    if (lane < 16) y[b * V + i0 + m] = yv;

    // Deterministic per-tile partial sums (lanes 16..31 mirror 0..15).
    float ps = yv, pq = yv * yv;
#pragma unroll
    for (int off = 1; off < 16; off <<= 1) {
        ps += __shfl_xor(ps, off, 32);
        pq += __shfl_xor(pq, off, 32);
    }
    if (lane == 0) {
        psum[b * NTILE + tile]   = ps;
        psumsq[b * NTILE + tile] = pq;
    }
}

// ---------------------------------------------------------------------------
// Per-batch: reduce 128 partials (fixed tree -> deterministic), normalize.
// grid NB, block 256
// ---------------------------------------------------------------------------
__global__ void __launch_bounds__(256)
sbm_norm(const float* __restrict__ y, const float* __restrict__ psum,
         const float* __restrict__ psumsq, float* __restrict__ xout) {
    const int b   = blockIdx.x;
    const int tid = threadIdx.x;
    __shared__ float ssum[NTILE];
    __shared__ float ssq[NTILE];
    if (tid < NTILE) {
        ssum[tid] = psum[b * NTILE + tid];
        ssq[tid]  = psumsq[b * NTILE + tid];
    }
    __syncthreads();
    for (int sft = NTILE / 2; sft > 0; sft >>= 1) {
        if (tid < sft) {
            ssum[tid] += ssum[tid + sft];
            ssq[tid]  += ssq[tid + sft];
        }
        __syncthreads();
    }
    const float mu  = ssum[0] * (1.0f / V);
    const float var = ssq[0] * (1.0f / V) - mu * mu;
    const float inv = rsqrtf(var + EPS);
    for (int i = tid; i < V; i += 256)
        xout[b * V + i] = (y[b * V + i] - mu) * inv;
}

// ---------------------------------------------------------------------------
extern "C" void kernel_launch(void* const* d_in, const int* in_sizes, int n_in,
                              void* d_out, int out_size, void* d_ws,
                              size_t ws_size, hipStream_t stream) {
    const float* A     = (const float*)d_in[0];
    const float* alpha = (const float*)d_in[1];
    const float* beta  = (const float*)d_in[2];
    float* out = (float*)d_out;

    float* xbuf   = (float*)d_ws;          // NB*V
    float* ybuf   = xbuf + NB * V;         // NB*V
    float* psum   = ybuf + NB * V;         // NB*NTILE
    float* psumsq = psum + NB * NTILE;     // NB*NTILE

    sbm_colsum<<<dim3(V / 64, NB), 64, 0, stream>>>(A, xbuf);

    for (int k = 0; k < NSTEP; ++k) {
        sbm_step<<<dim3(NTILE / 4, NB), 128, 0, stream>>>(
            A, xbuf, alpha, beta, k, ybuf, psum, psumsq);
        sbm_norm<<<dim3(NB), 256, 0, stream>>>(
            ybuf, psum, psumsq, (k == NSTEP - 1) ? out : xbuf);
    }
}